// GraphSAGE_74792560492685
// MI455X (gfx1250) — compile-verified
//
#include <hip/hip_runtime.h>
#include <hip/hip_bf16.h>

typedef __attribute__((ext_vector_type(2))) float v2f;
typedef __attribute__((ext_vector_type(8))) float v8f;

#define N_NODES 100000
#define N_EDGES 1600000
#define IN_F    128
#define HID_F   128
#define OUT_F   64

// ---------------------------------------------------------------- utilities
__global__ void zero_f32(float* __restrict__ p, int n) {
    int i = blockIdx.x * blockDim.x + threadIdx.x;
    if (i < n) p[i] = 0.0f;
}

__global__ void degree_kernel(const int* __restrict__ dst, float* __restrict__ deg, int n_edges) {
    int e = blockIdx.x * blockDim.x + threadIdx.x;
    if (e < n_edges) atomicAdd(&deg[dst[e]], 1.0f);
}

__global__ void invdeg_kernel(float* __restrict__ deg, int n) {
    int i = blockIdx.x * blockDim.x + threadIdx.x;
    if (i < n) deg[i] = 1.0f / fmaxf(deg[i], 1.0f);
}

// ------------------------------------------------------------- aggregation
// One wave (32 lanes) per edge; lane c moves features [4c, 4c+4).
// src/dst loads are wave-uniform; gather + atomics are 512B coalesced.
__global__ void aggregate_kernel(const float* __restrict__ h,
                                 const int*   __restrict__ src,
                                 const int*   __restrict__ dst,
                                 float*       __restrict__ agg,
                                 int n_edges) {
    long long t = (long long)blockIdx.x * blockDim.x + threadIdx.x;
    int e = (int)(t >> 5);
    int c = (int)(t & 31);
    if (e >= n_edges) return;
    int s = src[e];
    int d = dst[e];
    const float4 v = *(const float4*)(h + (long long)s * 128 + c * 4);
    float* ap = agg + (long long)d * 128 + c * 4;
    atomicAdd(ap + 0, v.x);
    atomicAdd(ap + 1, v.y);
    atomicAdd(ap + 2, v.z);
    atomicAdd(ap + 3, v.w);
}

// ----------------------------------------------------- fused WMMA SAGE GEMM
// out[row0..row0+16) = act( h@Ws + (agg * inv_deg)@Wn + b )
// One wave per 16x16 tile; threadIdx.y selects column tile (covers all NOUT).
// A fragment (16x4 f32, 2 VGPRs): lane L holds row M=L&15, K = koff, koff+1
//   with koff = (L>=16) ? 2 : 0 (ISA 7.12.2 layout).
// B fragment (4x16 f32, 2 VGPRs): lane L holds col N=L&15, rows K = koff, koff+1.
// C/D (8 VGPRs): VGPR i -> row i (lanes 0-15) / row i+8 (lanes 16-31), col = lane&15.
template <int NOUT, bool RELU>
__global__ __launch_bounds__(32 * (NOUT / 16))
void sage_gemm_wmma(const float* __restrict__ h,
                    const float* __restrict__ agg,
                    const float* __restrict__ inv_deg,
                    const float* __restrict__ Ws,
                    const float* __restrict__ Wn,
                    const float* __restrict__ bias,
                    float* __restrict__ out) {
    const int lane = threadIdx.x;                 // 0..31
    const int n0   = threadIdx.y * 16;            // column tile base
    const int row0 = blockIdx.x * 16;             // row tile base
    const int m    = lane & 15;                   // row (A) / col (B,C) within tile
    const int koff = (lane >> 4) << 1;            // 0 or 2

    const float* hrow = h   + (long long)(row0 + m) * 128;
    const float* arow = agg + (long long)(row0 + m) * 128;
    const float scale = inv_deg[row0 + m];

    v8f acc = {};
#pragma unroll
    for (int k = 0; k < 128; k += 4) {
        // self term: h @ Ws
        v2f a = *(const v2f*)(hrow + k + koff);
        v2f bw;
        bw.x = Ws[(k + koff + 0) * NOUT + n0 + m];
        bw.y = Ws[(k + koff + 1) * NOUT + n0 + m];
        acc = __builtin_amdgcn_wmma_f32_16x16x4_f32(false, a, false, bw,
                                                    (short)0, acc, false, false);
        // neighbor term: (agg * inv_deg) @ Wn
        v2f a2 = *(const v2f*)(arow + k + koff);
        a2.x *= scale;
        a2.y *= scale;
        v2f bn;
        bn.x = Wn[(k + koff + 0) * NOUT + n0 + m];
        bn.y = Wn[(k + koff + 1) * NOUT + n0 + m];
        acc = __builtin_amdgcn_wmma_f32_16x16x4_f32(false, a2, false, bn,
                                                    (short)0, acc, false, false);
    }

    // In-place safety (layer 1 writes over h): all waves in this block read
    // the same 16 rows above; make sure every read is done before any store.
    __syncthreads();

    const float bv   = bias[n0 + m];
    const int rbase  = row0 + ((lane >> 4) << 3);
#pragma unroll
    for (int i = 0; i < 8; ++i) {
        float v = acc[i] + bv;
        if (RELU) v = fmaxf(v, 0.0f);
        out[(long long)(rbase + i) * NOUT + n0 + m] = v;
    }
}

// ------------------------------------------------------------------ launch
extern "C" void kernel_launch(void* const* d_in, const int* in_sizes, int n_in,
                              void* d_out, int out_size, void* d_ws, size_t ws_size,
                              hipStream_t stream) {
    const float* feat = (const float*)d_in[0];
    const int*   src  = (const int*)d_in[1];
    const int*   dst  = (const int*)d_in[2];
    const float* Ws0  = (const float*)d_in[3];
    const float* Wn0  = (const float*)d_in[4];
    const float* b0   = (const float*)d_in[5];
    const float* Ws1  = (const float*)d_in[6];
    const float* Wn1  = (const float*)d_in[7];
    const float* b1   = (const float*)d_in[8];
    const float* Ws2  = (const float*)d_in[9];
    const float* Wn2  = (const float*)d_in[10];
    const float* b2   = (const float*)d_in[11];
    float* out = (float*)d_out;

    // workspace carve-up (512B-aligned sections)
    char* ws = (char*)d_ws;
    float* inv_deg = (float*)ws;                                   // 100000 f32
    size_t off_agg = ((size_t)N_NODES * 4 + 511) & ~(size_t)511;   // 400384
    float* agg = (float*)(ws + off_agg);                           // 100000*128 f32
    float* h   = agg + (size_t)N_NODES * HID_F;                    // 100000*128 f32

    const int NF = N_NODES * HID_F;                                // 12.8M
    const int agg_blocks = (int)(((long long)N_EDGES * 32 + 255) / 256);  // 200000
    const int row_tiles  = N_NODES / 16;                           // 6250 exact
    dim3 blk128(32, 8), blk64(32, 4);

    // degrees -> inv_deg (in place)
    zero_f32      <<<(N_NODES + 255) / 256, 256, 0, stream>>>(inv_deg, N_NODES);
    degree_kernel <<<(N_EDGES + 255) / 256, 256, 0, stream>>>(dst, inv_deg, N_EDGES);
    invdeg_kernel <<<(N_NODES + 255) / 256, 256, 0, stream>>>(inv_deg, N_NODES);

    // ---- layer 0: feat -> h
    zero_f32        <<<(NF + 255) / 256, 256, 0, stream>>>(agg, NF);
    aggregate_kernel<<<agg_blocks, 256, 0, stream>>>(feat, src, dst, agg, N_EDGES);
    sage_gemm_wmma<128, true><<<row_tiles, blk128, 0, stream>>>(feat, agg, inv_deg,
                                                                Ws0, Wn0, b0, h);

    // ---- layer 1: h -> h (in place)
    zero_f32        <<<(NF + 255) / 256, 256, 0, stream>>>(agg, NF);
    aggregate_kernel<<<agg_blocks, 256, 0, stream>>>(h, src, dst, agg, N_EDGES);
    sage_gemm_wmma<128, true><<<row_tiles, blk128, 0, stream>>>(h, agg, inv_deg,
                                                                Ws1, Wn1, b1, h);

    // ---- layer 2: h -> out (no relu, NOUT=64)
    zero_f32        <<<(NF + 255) / 256, 256, 0, stream>>>(agg, NF);
    aggregate_kernel<<<agg_blocks, 256, 0, stream>>>(h, src, dst, agg, N_EDGES);
    sage_gemm_wmma<64, false><<<row_tiles, blk64, 0, stream>>>(h, agg, inv_deg,
                                                               Ws2, Wn2, b2, out);
}